// RANSACBlock_4337916969609
// MI455X (gfx1250) — compile-verified
//
#include <hip/hip_runtime.h>

#define BATCH 64
#define NPTS  16384
#define NSAMP 6
#define NIT   256
#define THR2  0.25f   // (ERROR_TOLERANCE=0.5)^2
#define TPW   4       // point tiles (16 pts) per wave in the scoring kernel

typedef __attribute__((ext_vector_type(2))) float v2f;
typedef __attribute__((ext_vector_type(8))) float v8f;

// ---------------------------------------------------------------------------
// Threefry-2x32, JAX-compatible (PRNGKey(42) -> key = {0, 42}).
// jax.random.randint(key,(256,64,6),0,16384): counts = iota(98304) split in
// halves, 20 rounds, result & 16383 (JAX randint multiplier is 0 for span 2^14).
// ---------------------------------------------------------------------------
__device__ __forceinline__ unsigned tf_rotl(unsigned x, int r) {
  return (x << r) | (x >> (32 - r));
}

__device__ unsigned threefry2x32(unsigned c0, unsigned c1, int second) {
  const unsigned k0 = 0u, k1 = 42u, k2 = 0u ^ 42u ^ 0x1BD11BDAu;
  unsigned x0 = c0 + k0, x1 = c1 + k1;
#define TF_R(r) { x0 += x1; x1 = tf_rotl(x1, r); x1 ^= x0; }
#define TF_A TF_R(13) TF_R(15) TF_R(26) TF_R(6)
#define TF_B TF_R(17) TF_R(29) TF_R(16) TF_R(24)
  TF_A x0 += k1; x1 += k2 + 1u;
  TF_B x0 += k2; x1 += k0 + 2u;
  TF_A x0 += k0; x1 += k1 + 3u;
  TF_B x0 += k1; x1 += k2 + 4u;
  TF_A x0 += k2; x1 += k0 + 5u;
#undef TF_R
#undef TF_A
#undef TF_B
  return second ? x1 : x0;
}

__device__ __forceinline__ float det3(const float M[3][3]) {
  return M[0][0] * (M[1][1] * M[2][2] - M[1][2] * M[2][1])
       - M[0][1] * (M[1][0] * M[2][2] - M[1][2] * M[2][0])
       + M[0][2] * (M[1][0] * M[2][1] - M[1][1] * M[2][0]);
}

__device__ void mat4mul(const float* A, const float* B, float* C) {
#pragma unroll
  for (int i = 0; i < 4; ++i)
#pragma unroll
    for (int j = 0; j < 4; ++j) {
      float s = 0.0f;
#pragma unroll
      for (int k = 0; k < 4; ++k) s += A[i * 4 + k] * B[k * 4 + j];
      C[i * 4 + j] = s;
    }
}

// ---------------------------------------------------------------------------
// Kernel 1: one thread per (iteration, batch) hypothesis.
// Produces per-hypothesis weight vector w[16] such that
//   err^2(point) = (|src|^2+|trg|^2) + dot(w, f(point)),  f = [src,trg,trg(x)src,1]
// Stored twice: plain [b][it][16] and pre-swizzled into WMMA A-matrix layout.
// ---------------------------------------------------------------------------
__global__ void ransac_hypo_kernel(const float* __restrict__ src,
                                   const float* __restrict__ trg,
                                   const float* __restrict__ wts,
                                   const float* __restrict__ Tsv,
                                   float* __restrict__ wbuf,
                                   float* __restrict__ wplain) {
  const int gid = blockIdx.x * blockDim.x + threadIdx.x;
  if (gid >= NIT * BATCH) return;
  const int it = gid >> 6;          // rand_idx is (iter, batch, samp)
  const int b  = gid & (BATCH - 1);

  // ---- sample indices via threefry ----
  const unsigned HALF = (unsigned)(NIT * BATCH * NSAMP) / 2u;  // 49152
  int idx[NSAMP];
#pragma unroll
  for (int s = 0; s < NSAMP; ++s) {
    const unsigned i = (unsigned)gid * NSAMP + (unsigned)s;
    const unsigned j = (i < HALF) ? i : (i - HALF);
    idx[s] = (int)(threefry2x32(j, j + HALF, i >= HALF) & (unsigned)(NPTS - 1));
  }

  // ---- gather samples ----
  float ps[3][NSAMP], pt[3][NSAMP], sw[NSAMP];
  const size_t pb = (size_t)b * 4 * NPTS;
  const size_t ob = (size_t)b * NPTS;
#pragma unroll
  for (int s = 0; s < NSAMP; ++s) {
    const int n = idx[s];
#pragma unroll
    for (int c = 0; c < 3; ++c) {
      ps[c][s] = src[pb + (size_t)c * NPTS + n];
      pt[c][s] = trg[pb + (size_t)c * NPTS + n];
    }
    sw[s] = wts[ob + n];
  }

  // ---- weighted centroids ----
  float wsum = 0.0f;
#pragma unroll
  for (int s = 0; s < NSAMP; ++s) wsum += sw[s];
  const float winv = 1.0f / wsum;
  float cs[3] = {0, 0, 0}, ctn[3] = {0, 0, 0};
#pragma unroll
  for (int s = 0; s < NSAMP; ++s)
#pragma unroll
    for (int c = 0; c < 3; ++c) {
      cs[c]  += ps[c][s] * sw[s];
      ctn[c] += pt[c][s] * sw[s];
    }
#pragma unroll
  for (int c = 0; c < 3; ++c) { cs[c] *= winv; ctn[c] *= winv; }

  // ---- weighted covariance cov[i][j] = sum_s (qt_i * w_s) * qs_j ----
  float cov[3][3] = {{0,0,0},{0,0,0},{0,0,0}};
#pragma unroll
  for (int s = 0; s < NSAMP; ++s) {
    float qt0 = pt[0][s] - ctn[0], qt1 = pt[1][s] - ctn[1], qt2 = pt[2][s] - ctn[2];
    float q0 = (ps[0][s] - cs[0]) * sw[s];
    float q1 = (ps[1][s] - cs[1]) * sw[s];
    float q2 = (ps[2][s] - cs[2]) * sw[s];
    cov[0][0] += qt0 * q0; cov[0][1] += qt0 * q1; cov[0][2] += qt0 * q2;
    cov[1][0] += qt1 * q0; cov[1][1] += qt1 * q1; cov[1][2] += qt1 * q2;
    cov[2][0] += qt2 * q0; cov[2][1] += qt2 * q1; cov[2][2] += qt2 * q2;
  }

  // ---- SVD via Jacobi eigendecomposition of covT*cov ----
  float Am[3][3], V[3][3] = {{1,0,0},{0,1,0},{0,0,1}};
#pragma unroll
  for (int i = 0; i < 3; ++i)
#pragma unroll
    for (int j = 0; j < 3; ++j)
      Am[i][j] = cov[0][i] * cov[0][j] + cov[1][i] * cov[1][j] + cov[2][i] * cov[2][j];

  const int PP[3] = {0, 0, 1}, QQ[3] = {1, 2, 2};
  for (int sweep = 0; sweep < 10; ++sweep) {
#pragma unroll
    for (int pi = 0; pi < 3; ++pi) {
      const int p = PP[pi], q = QQ[pi];
      const float apq = Am[p][q];
      if (fabsf(apq) > 1e-18f) {
        const float theta = (Am[q][q] - Am[p][p]) / (2.0f * apq);
        const float tt = copysignf(1.0f, theta) / (fabsf(theta) + sqrtf(theta * theta + 1.0f));
        const float cc = rsqrtf(tt * tt + 1.0f);
        const float ss = tt * cc;
        const float app = Am[p][p], aqq = Am[q][q];
        Am[p][p] = app - tt * apq;
        Am[q][q] = aqq + tt * apq;
        Am[p][q] = Am[q][p] = 0.0f;
        const int r = 3 - p - q;
        const float arp = Am[r][p], arq = Am[r][q];
        Am[r][p] = Am[p][r] = cc * arp - ss * arq;
        Am[r][q] = Am[q][r] = ss * arp + cc * arq;
#pragma unroll
        for (int k = 0; k < 3; ++k) {
          const float vkp = V[k][p], vkq = V[k][q];
          V[k][p] = cc * vkp - ss * vkq;
          V[k][q] = ss * vkp + cc * vkq;
        }
      }
    }
  }

  // sort eigenpairs descending (match svd convention)
  float lam[3] = {Am[0][0], Am[1][1], Am[2][2]};
#pragma unroll
  for (int a2 = 0; a2 < 2; ++a2)
#pragma unroll
    for (int b2 = 1; b2 < 3; ++b2)
      if (b2 > a2 && lam[b2] > lam[a2]) {
        float tl = lam[a2]; lam[a2] = lam[b2]; lam[b2] = tl;
#pragma unroll
        for (int k = 0; k < 3; ++k) { float tv = V[k][a2]; V[k][a2] = V[k][b2]; V[k][b2] = tv; }
      }

  // left singular vectors U = cov * V / sigma (with degenerate fallbacks)
  float U[3][3];
  int bad[3];
  const float smax = sqrtf(fmaxf(lam[0], 0.0f));
#pragma unroll
  for (int j = 0; j < 3; ++j) {
    float u0 = cov[0][0] * V[0][j] + cov[0][1] * V[1][j] + cov[0][2] * V[2][j];
    float u1 = cov[1][0] * V[0][j] + cov[1][1] * V[1][j] + cov[1][2] * V[2][j];
    float u2 = cov[2][0] * V[0][j] + cov[2][1] * V[1][j] + cov[2][2] * V[2][j];
    const float nn = sqrtf(u0 * u0 + u1 * u1 + u2 * u2);
    if (nn > fmaxf(1e-6f * smax, 1e-20f)) {
      const float rn = 1.0f / nn;
      U[0][j] = u0 * rn; U[1][j] = u1 * rn; U[2][j] = u2 * rn;
      bad[j] = 0;
    } else {
      U[0][j] = 0.0f; U[1][j] = 0.0f; U[2][j] = 0.0f;
      bad[j] = 1;
    }
  }
  if (bad[0]) {
    U[0][0] = 1; U[1][0] = 0; U[2][0] = 0;
    U[0][1] = 0; U[1][1] = 1; U[2][1] = 0;
    U[0][2] = 0; U[1][2] = 0; U[2][2] = 1;
  } else {
    if (bad[1]) {
      const float ex = (fabsf(U[0][0]) < 0.9f) ? 1.0f : 0.0f;
      const float ey = 1.0f - ex;
      float c0 = -U[2][0] * ey;
      float c1 =  U[2][0] * ex;
      float c2 =  U[0][0] * ey - U[1][0] * ex;
      const float rn = rsqrtf(fmaxf(c0 * c0 + c1 * c1 + c2 * c2, 1e-30f));
      U[0][1] = c0 * rn; U[1][1] = c1 * rn; U[2][1] = c2 * rn;
    }
    if (bad[2]) {
      U[0][2] = U[1][0] * U[2][1] - U[2][0] * U[1][1];
      U[1][2] = U[2][0] * U[0][1] - U[0][0] * U[2][1];
      U[2][2] = U[0][0] * U[1][1] - U[1][0] * U[0][1];
    }
  }

  const float dd = (det3(U) * det3(V) < 0.0f) ? -1.0f : 1.0f;
  float R[3][3];
#pragma unroll
  for (int i = 0; i < 3; ++i)
#pragma unroll
    for (int j = 0; j < 3; ++j)
      R[i][j] = U[i][0] * V[j][0] + U[i][1] * V[j][1] + dd * U[i][2] * V[j][2];

  float tv3[3];
#pragma unroll
  for (int i = 0; i < 3; ++i)
    tv3[i] = ctn[i] - (R[i][0] * cs[0] + R[i][1] * cs[1] + R[i][2] * cs[2]);

  // T_cam = T_s_v * T * inv(T_s_v)
  float Tm[16] = {R[0][0], R[0][1], R[0][2], tv3[0],
                  R[1][0], R[1][1], R[1][2], tv3[1],
                  R[2][0], R[2][1], R[2][2], tv3[2],
                  0, 0, 0, 1};
  float Ts[16];
#pragma unroll
  for (int k = 0; k < 16; ++k) Ts[k] = Tsv[k];
  float Tv[16];
#pragma unroll
  for (int i = 0; i < 3; ++i) {
#pragma unroll
    for (int j = 0; j < 3; ++j) Tv[i * 4 + j] = Ts[j * 4 + i];
    Tv[i * 4 + 3] = -(Ts[0 * 4 + i] * Ts[3] + Ts[1 * 4 + i] * Ts[7] + Ts[2 * 4 + i] * Ts[11]);
  }
  Tv[12] = 0; Tv[13] = 0; Tv[14] = 0; Tv[15] = 1;
  float T1[16], Tc[16];
  mat4mul(Ts, Tm, T1);
  mat4mul(T1, Tv, Tc);

  float Rc[3][3], tc[3];
#pragma unroll
  for (int i = 0; i < 3; ++i) {
#pragma unroll
    for (int j = 0; j < 3; ++j) Rc[i][j] = Tc[i * 4 + j];
    tc[i] = Tc[i * 4 + 3];
  }

  // ---- hypothesis weight vector ----
  float wv[16];
#pragma unroll
  for (int j = 0; j < 3; ++j)
    wv[j] = 2.0f * (Rc[0][j] * tc[0] + Rc[1][j] * tc[1] + Rc[2][j] * tc[2]);
#pragma unroll
  for (int i = 0; i < 3; ++i) wv[3 + i] = -2.0f * tc[i];
#pragma unroll
  for (int i = 0; i < 3; ++i)
#pragma unroll
    for (int j = 0; j < 3; ++j) wv[6 + 3 * i + j] = -2.0f * Rc[i][j];
  wv[15] = tc[0] * tc[0] + tc[1] * tc[1] + tc[2] * tc[2];

  float* wpl = wplain + ((size_t)b * NIT + it) * 16;
#pragma unroll
  for (int d = 0; d < 16; ++d) wpl[d] = wv[d];

  // A-matrix (16x4 f32) lane layout: VGPR0 = K0 (lanes 0-15) / K2 (16-31),
  // VGPR1 = K1 / K3.  reg r = kc*2 + (c&1), lane = m + 16*(c>>1), dim = kc*4+c.
  const int ht = it >> 4, m = it & 15;
  float* wbb = wbuf + ((size_t)b * 16 + ht) * 32 * 8;
#pragma unroll
  for (int d = 0; d < 16; ++d) {
    const int c = d & 3, kc = d >> 2;
    const int lane = m + ((c >> 1) << 4);
    const int r = kc * 2 + (c & 1);
    wbb[lane * 8 + r] = wv[d];
  }
}

// ---------------------------------------------------------------------------
// Kernel 2: WMMA scoring.  Workgroup = 1 batch x 512 points (32 tiles, 8 waves
// x TPW=4 tiles), loops all 256 hypotheses (16 hyp-tiles).  err^2 tile
// (16 hyp x 16 pts) = 4 chained V_WMMA_F32_16X16X4_F32 with C preloaded with
// per-point |src|^2+|trg|^2.  Counting is branchless: ballot -> uniform SALU
// popcounts -> cndmask select-tree scatter -> single full-exec LDS atomic per
// hyp-tile.
// ---------------------------------------------------------------------------
__global__ void __launch_bounds__(256) ransac_score_kernel(
    const float* __restrict__ src, const float* __restrict__ trg,
    const int* __restrict__ vsrc, const int* __restrict__ vtrg,
    const float* __restrict__ wbuf, int* __restrict__ counts) {
  const int b     = blockIdx.y;
  const int chunk = blockIdx.x;            // 32 chunks of 512 points
  const int tid   = threadIdx.x;
  const int wave  = tid >> 5;
  const int lane  = tid & 31;
  const int hi    = lane >> 4;
  const int ln    = lane & 15;
  const bool L0   = (lane & 1) != 0;       // lane bits for select tree
  const bool L1   = (lane & 2) != 0;
  const bool L2   = (lane & 4) != 0;
  const bool L3   = (lane & 8) != 0;
  const bool LLO  = lane < 16;

  __shared__ float sW[16 * 32 * 8];        // all hypothesis weights (A layout), 16 KB
  __shared__ int   sCnt[NIT];

  sCnt[tid] = 0;
  {
    const float* gw = wbuf + (size_t)b * (16 * 32 * 8);
    for (int u = tid; u < 16 * 32 * 8; u += 256) sW[u] = gw[u];
  }
  __syncthreads();

  // Build B-matrices (features) for this wave's TPW point tiles, in registers.
  // B (4x16 f32) layout: VGPR0 = K0 (lanes 0-15) / K1 (16-31), VGPR1 = K2/K3,
  // N = lane%16  =>  Bm[r] = f[2*r + hi].
  float    Bm[TPW][8];
  float    Cc[TPW];
  unsigned vb[TPW];                        // uniform ballot of valid bits
  const size_t pb = (size_t)b * 4 * NPTS;
  const size_t vbo = (size_t)b * NPTS;
#pragma unroll
  for (int t = 0; t < TPW; ++t) {
    const int ptid = (chunk * (8 * TPW) + wave * TPW + t) * 16 + ln;
    const float sx = src[pb + ptid];
    const float sy = src[pb + NPTS + ptid];
    const float sz = src[pb + 2 * NPTS + ptid];
    const float tx = trg[pb + ptid];
    const float ty = trg[pb + NPTS + ptid];
    const float tz = trg[pb + 2 * NPTS + ptid];
    const float f[16] = {sx, sy, sz, tx, ty, tz,
                         tx * sx, tx * sy, tx * sz,
                         ty * sx, ty * sy, ty * sz,
                         tz * sx, tz * sy, tz * sz, 1.0f};
#pragma unroll
    for (int r = 0; r < 8; ++r) Bm[t][r] = f[2 * r + hi];
    Cc[t] = sx * sx + sy * sy + sz * sz + tx * tx + ty * ty + tz * tz;
    const bool val = (vsrc[vbo + ptid] > 0) && (vtrg[vbo + ptid] > 0);
    vb[t] = __builtin_amdgcn_ballot_w32(val);
  }

  for (int ht = 0; ht < 16; ++ht) {
    float a[8];
    const float* wp = &sW[(ht * 32 + lane) * 8];
#pragma unroll
    for (int r = 0; r < 8; ++r) a[r] = wp[r];

    unsigned cl[8], ch[8];                 // uniform per-wave counts, hyp r / r+8
#pragma unroll
    for (int r = 0; r < 8; ++r) { cl[r] = 0u; ch[r] = 0u; }

#pragma unroll
    for (int t = 0; t < TPW; ++t) {
      v8f d;
#pragma unroll
      for (int r = 0; r < 8; ++r) d[r] = Cc[t];
#pragma unroll
      for (int kc = 0; kc < 4; ++kc) {
        v2f av, bv;
        av[0] = a[kc * 2];     av[1] = a[kc * 2 + 1];
        bv[0] = Bm[t][kc * 2]; bv[1] = Bm[t][kc * 2 + 1];
        d = __builtin_amdgcn_wmma_f32_16x16x4_f32(
            /*neg_a=*/false, av, /*neg_b=*/false, bv,
            /*c_mod=*/(short)0, d, /*reuse_a=*/false, /*reuse_b=*/false);
      }
      // D layout: lane l, VGPR r -> hyp M = r + 8*(l>=16), point N = l%16.
#pragma unroll
      for (int r = 0; r < 8; ++r) {
        const unsigned m = __builtin_amdgcn_ballot_w32(d[r] < THR2) & vb[t];
        cl[r] += (unsigned)__popc(m & 0xFFFFu);
        ch[r] += (unsigned)__popc(m >> 16);
      }
    }

    // Scatter 16 uniform counts to lanes 0..15 via a branchless cndmask
    // select tree (lane i gets count for hyp ht*16+i; lanes 16..31 get 0),
    // then ONE full-exec LDS atomic.
    const unsigned a0 = L0 ? cl[1] : cl[0];
    const unsigned a1 = L0 ? cl[3] : cl[2];
    const unsigned a2 = L0 ? cl[5] : cl[4];
    const unsigned a3 = L0 ? cl[7] : cl[6];
    const unsigned a4 = L0 ? ch[1] : ch[0];
    const unsigned a5 = L0 ? ch[3] : ch[2];
    const unsigned a6 = L0 ? ch[5] : ch[4];
    const unsigned a7 = L0 ? ch[7] : ch[6];
    const unsigned b0 = L1 ? a1 : a0;
    const unsigned b1 = L1 ? a3 : a2;
    const unsigned b2 = L1 ? a5 : a4;
    const unsigned b3 = L1 ? a7 : a6;
    const unsigned c0 = L2 ? b1 : b0;
    const unsigned c1 = L2 ? b3 : b2;
    const unsigned dv = L3 ? c1 : c0;
    const int v = LLO ? (int)dv : 0;
    atomicAdd(&sCnt[ht * 16 + ln], v);     // lanes 16-31 add 0
  }
  __syncthreads();
  atomicAdd(&counts[b * NIT + tid], sCnt[tid]);
}

// ---------------------------------------------------------------------------
// Kernel 3: per-batch argmax with earliest-iteration tie-break; -1 if max==0
// (reference keeps all-false initial mask when no hypothesis beats 0).
// ---------------------------------------------------------------------------
__global__ void ransac_select_kernel(const int* __restrict__ counts,
                                     int* __restrict__ best) {
  const int b = blockIdx.x;
  const int t = threadIdx.x;
  __shared__ int sc[NIT];
  __shared__ int si[NIT];
  sc[t] = counts[b * NIT + t];
  si[t] = t;
  __syncthreads();
  for (int off = NIT / 2; off > 0; off >>= 1) {
    if (t < off) {
      const int c2 = sc[t + off], i2 = si[t + off];
      if (c2 > sc[t] || (c2 == sc[t] && i2 < si[t])) { sc[t] = c2; si[t] = i2; }
    }
    __syncthreads();
  }
  if (t == 0) best[b] = (sc[0] > 0) ? si[0] : -1;
}

// ---------------------------------------------------------------------------
// Kernel 4: emit inlier mask for the winning hypothesis.
// ---------------------------------------------------------------------------
__global__ void ransac_emit_kernel(const float* __restrict__ src,
                                   const float* __restrict__ trg,
                                   const int* __restrict__ vsrc,
                                   const int* __restrict__ vtrg,
                                   const float* __restrict__ wplain,
                                   const int* __restrict__ best,
                                   float* __restrict__ out) {
  const int b = blockIdx.y;
  const int n = blockIdx.x * blockDim.x + threadIdx.x;
  const int h = best[b];
  float o = 0.0f;
  if (h >= 0) {
    const float* w = wplain + ((size_t)b * NIT + h) * 16;
    const size_t pb = (size_t)b * 4 * NPTS;
    const float sx = src[pb + n], sy = src[pb + NPTS + n], sz = src[pb + 2 * NPTS + n];
    const float tx = trg[pb + n], ty = trg[pb + NPTS + n], tz = trg[pb + 2 * NPTS + n];
    float e = sx * sx + sy * sy + sz * sz + tx * tx + ty * ty + tz * tz;
    e += w[0] * sx + w[1] * sy + w[2] * sz;
    e += w[3] * tx + w[4] * ty + w[5] * tz;
    e += w[6] * (tx * sx) + w[7] * (tx * sy) + w[8] * (tx * sz);
    e += w[9] * (ty * sx) + w[10] * (ty * sy) + w[11] * (ty * sz);
    e += w[12] * (tz * sx) + w[13] * (tz * sy) + w[14] * (tz * sz);
    e += w[15];
    const bool val = (vsrc[(size_t)b * NPTS + n] > 0) && (vtrg[(size_t)b * NPTS + n] > 0);
    o = (e < THR2 && val) ? 1.0f : 0.0f;
  }
  out[(size_t)b * NPTS + n] = o;
}

// ---------------------------------------------------------------------------
extern "C" void kernel_launch(void* const* d_in, const int* in_sizes, int n_in,
                              void* d_out, int out_size, void* d_ws, size_t ws_size,
                              hipStream_t stream) {
  const float* src = (const float*)d_in[0];   // keypoints_3D_src (64,4,16384)
  const float* trg = (const float*)d_in[1];   // keypoints_3D_trg (64,4,16384)
  // d_in[2] = keypoints_2D_trg : unused (DIM == '3D')
  const int*   vs  = (const int*)d_in[3];     // valid_pts_src (64,1,16384)
  const int*   vt  = (const int*)d_in[4];     // valid_pts_trg (64,1,16384)
  const float* wts = (const float*)d_in[5];   // weights (64,1,16384)
  const float* Tsv = (const float*)d_in[6];   // T_s_v (4,4)

  // workspace carve-out (~2.2 MB)
  float* wbuf   = (float*)d_ws;                                   // 64*16*32*8 f
  float* wplain = wbuf + (size_t)BATCH * 16 * 32 * 8;             // 64*256*16 f
  int*   counts = (int*)(wplain + (size_t)BATCH * NIT * 16);      // 64*256 i
  int*   best   = counts + BATCH * NIT;                           // 64 i

  (void)hipMemsetAsync(counts, 0, BATCH * NIT * sizeof(int), stream);

  ransac_hypo_kernel<<<dim3((NIT * BATCH) / 256), 256, 0, stream>>>(
      src, trg, wts, Tsv, wbuf, wplain);

  ransac_score_kernel<<<dim3(NPTS / (8 * TPW * 16), BATCH), 256, 0, stream>>>(
      src, trg, vs, vt, wbuf, counts);

  ransac_select_kernel<<<dim3(BATCH), NIT, 0, stream>>>(counts, best);

  ransac_emit_kernel<<<dim3(NPTS / 256, BATCH), 256, 0, stream>>>(
      src, trg, vs, vt, wplain, best, (float*)d_out);
}